// MultiHeadAttention_4784593568493
// MI455X (gfx1250) — compile-verified
//
#include <hip/hip_runtime.h>
#include <math.h>

#define L 32768
#define E 1024
#define H 8
#define D 128

typedef __attribute__((ext_vector_type(2))) float v2f;
typedef __attribute__((ext_vector_type(8))) float v8f;

// ---------------------------------------------------------------------------
// wave32 reduction
__device__ __forceinline__ float waveSum(float v) {
    for (int off = 16; off > 0; off >>= 1) v += __shfl_down(v, off, 32);
    return v;
}

// ---------------------------------------------------------------------------
// zero a float buffer
__global__ void zeroK(float* p, int n) {
    int i = blockIdx.x * blockDim.x + threadIdx.x;
    if (i < n) p[i] = 0.0f;
}

// ---------------------------------------------------------------------------
// q[o] = bq[o] + sum_j x0[j] * Wq[o*E + j]      (one wave per output row)
__global__ void qprojK(const float* __restrict__ x, const float* __restrict__ W,
                       const float* __restrict__ bias, float* __restrict__ q) {
    int gw   = (blockIdx.x * blockDim.x + threadIdx.x) >> 5;   // 1024 waves
    int lane = threadIdx.x & 31;
    const float* wr = W + (size_t)gw * E;
    float s = 0.0f;
    for (int j = lane; j < E; j += 32) s += x[j] * wr[j];
    s = waveSum(s);
    if (lane == 0) q[gw] = s + bias[gw];
}

// ---------------------------------------------------------------------------
// qkT[e][h] = sum_d Wk[(E + h*128 + d)*E + e] * q[h*128 + d]   (pad h>=8 -> 0)
// consts[h] = q_h . bk_h
__global__ void qkK(const float* __restrict__ W, const float* __restrict__ bias,
                    const float* __restrict__ q, float* __restrict__ qkT,
                    float* __restrict__ consts) {
    __shared__ float qs[E];
    int tid = threadIdx.x;
    for (int i = tid; i < E; i += 256) qs[i] = q[i];
    __syncthreads();

    int e = blockIdx.x * 256 + tid;                    // 4 blocks x 256 = 1024
    for (int h = 0; h < H; ++h) {
        const float* wc = W + (size_t)(E + h * D) * E + e;
        float a = 0.0f;
        for (int d = 0; d < D; ++d) a += wc[(size_t)d * E] * qs[h * D + d];
        qkT[(size_t)e * 16 + h] = a;
    }
    for (int h = H; h < 16; ++h) qkT[(size_t)e * 16 + h] = 0.0f;

    if (blockIdx.x == 0 && tid < H) {
        float c = 0.0f;
        for (int d = 0; d < D; ++d) c += qs[tid * D + d] * bias[E + tid * D + d];
        consts[tid] = c;
    }
}

// ---------------------------------------------------------------------------
// Pass 1: scores[h][l] = (x_l . qk_h + const_h) * rsqrt(D)
// WMMA f32 16x16x4, x staged via async global->LDS (double buffered).
#define KC     32                 // k-chunk
#define RSTR   36                 // LDS row stride (16B aligned, conflict-free)
__global__ void __launch_bounds__(256) scoresK(const float* __restrict__ x,
                                               const float* __restrict__ qkT,
                                               const float* __restrict__ consts,
                                               float* __restrict__ scores) {
    __shared__ __align__(16) float tile[2][128 * RSTR];
    int tid  = threadIdx.x;
    int lane = tid & 31;
    int wave = tid >> 5;
    int half = lane >> 4;                              // 0 or 1
    int l15  = lane & 15;
    int base = blockIdx.x * 128;

    // this thread's staging slot: row r0(+32s), cols c4..c4+3 (b128, 16B aligned)
    int r0 = tid >> 3;
    int c4 = (tid & 7) * 4;

    v8f acc; for (int i = 0; i < 8; ++i) acc[i] = 0.0f;

    // issue 4 async b128 loads for k-chunk k0 into buffer b (per wave: ASYNCcnt +=4)
#define ISSUE(b, k0)                                                          \
    for (int s = 0; s < 4; ++s) {                                             \
        int r = r0 + s * 32;                                                  \
        unsigned long long ga =                                               \
            (unsigned long long)(size_t)(x + (size_t)(base + r) * E + (k0) + c4); \
        unsigned lo = (unsigned)(size_t)&tile[b][r * RSTR + c4];              \
        asm volatile("global_load_async_to_lds_b128 %0, %1, off"              \
                     :: "v"(lo), "v"(ga) : "memory");                         \
    }

    ISSUE(0, 0)
    for (int i = 0; i < E / KC; ++i) {                 // 32 chunks
        int k0 = i * KC;
        if (i + 1 < E / KC) {
            ISSUE((i + 1) & 1, k0 + KC)                // prefetch next chunk
            asm volatile("s_wait_asynccnt 0x4" ::: "memory");   // chunk i done
        } else {
            asm volatile("s_wait_asynccnt 0x0" ::: "memory");
        }
        __syncthreads();                               // all waves' data in LDS

        const float* tb = tile[i & 1];
        int rb = (wave * 16 + l15) * RSTR;
        for (int kk = 0; kk < KC; kk += 4) {
            // A (16x4): lanes 0-15 K={0,1}, lanes 16-31 K={2,3}
            int ka = kk + 2 * half;
            v2f a; a.x = tb[rb + ka]; a.y = tb[rb + ka + 1];
            // B (4x16): VGPR0 rows K=0/K=2 per half, VGPR1 rows K=1/K=3
            const float* bp = qkT + (size_t)(k0 + kk + 2 * half) * 16 + l15;
            v2f b; b.x = bp[0]; b.y = bp[16];
            acc = __builtin_amdgcn_wmma_f32_16x16x4_f32(
                false, a, false, b, (short)0, acc, false, false);
        }
        __syncthreads();                               // buffer reusable at i+2
    }
#undef ISSUE

    if (l15 < H) {                                     // valid head columns only
        float cst = consts[l15];
        for (int i = 0; i < 8; ++i) {
            int m   = i + 8 * half;                    // C: VGPR i -> M=i / i+8
            int row = base + wave * 16 + m;
            scores[(size_t)l15 * L + row] = (acc[i] + cst) * 0.08838834764831845f;
        }
    }
}

// ---------------------------------------------------------------------------
// Per-head softmax: attnT[l][h] = exp(s-max)/sum   (pad cols pre-zeroed)
__global__ void softmaxK(const float* __restrict__ scores, float* __restrict__ attnT) {
    int h = blockIdx.x, tid = threadIdx.x;             // 8 blocks x 256
    __shared__ float red[256];
    const float* s = scores + (size_t)h * L;

    float m = -INFINITY;
    for (int i = tid; i < L; i += 256) m = fmaxf(m, s[i]);
    red[tid] = m; __syncthreads();
    for (int st = 128; st > 0; st >>= 1) {
        if (tid < st) red[tid] = fmaxf(red[tid], red[tid + st]);
        __syncthreads();
    }
    m = red[0]; __syncthreads();

    float sum = 0.0f;
    for (int i = tid; i < L; i += 256) sum += __expf(s[i] - m);
    red[tid] = sum; __syncthreads();
    for (int st = 128; st > 0; st >>= 1) {
        if (tid < st) red[tid] += red[tid + st];
        __syncthreads();
    }
    float inv = 1.0f / red[0];
    for (int i = tid; i < L; i += 256)
        attnT[(size_t)i * 16 + h] = __expf(s[i] - m) * inv;
}

// ---------------------------------------------------------------------------
// Pass 2: ctx[h][e] += sum_l attn[h][l] * x[l][e]
// WMMA: A = attn tile (M=16 heads incl pad, K=4 l's), B = x rows (K=4 x N=16 e's).
__global__ void __launch_bounds__(256) ctxK(const float* __restrict__ x,
                                            const float* __restrict__ attnT,
                                            float* __restrict__ ctx) {
    int tid  = threadIdx.x;
    int lane = tid & 31;
    int wave = tid >> 5;
    int half = lane >> 4;
    int l15  = lane & 15;
    int l0   = blockIdx.x * 128;
    int e0   = wave * 128;

    v8f acc[8];
    for (int t = 0; t < 8; ++t)
        for (int i = 0; i < 8; ++i) acc[t][i] = 0.0f;

    for (int ks = 0; ks < 128; ks += 4) {
        int lk = l0 + ks + 2 * half;                   // A/B K rows {0,1}/{2,3} per half
        v2f a;
        a.x = attnT[(size_t)lk * 16 + l15];
        a.y = attnT[(size_t)(lk + 1) * 16 + l15];
        for (int t = 0; t < 8; ++t) {
            const float* bp = x + (size_t)lk * E + e0 + t * 16 + l15;
            __builtin_prefetch(bp + 4 * E, 0, 1);      // next l-chunk of this stream
            v2f b; b.x = bp[0]; b.y = bp[E];
            acc[t] = __builtin_amdgcn_wmma_f32_16x16x4_f32(
                false, a, false, b, (short)0, acc[t], false, false);
        }
    }

    if (half == 0) {                                   // C rows M=0..7 = real heads
        for (int t = 0; t < 8; ++t)
            for (int i = 0; i < 8; ++i)
                atomicAdd(&ctx[(size_t)i * E + e0 + t * 16 + l15], acc[t][i]);
    }
}

// ---------------------------------------------------------------------------
// vctx[o] = bv[o] + W_v[o] . ctx[o/128]          (one wave per output)
__global__ void vctxK(const float* __restrict__ W, const float* __restrict__ bias,
                      const float* __restrict__ ctx, float* __restrict__ vctx) {
    int gw   = (blockIdx.x * blockDim.x + threadIdx.x) >> 5;   // 1024 waves
    int lane = threadIdx.x & 31;
    const float* wr = W + (size_t)(2 * E + gw) * E;
    const float* cr = ctx + (size_t)(gw >> 7) * E;
    float s = 0.0f;
    for (int j = lane; j < E; j += 32) s += wr[j] * cr[j];
    s = waveSum(s);
    if (lane == 0) vctx[gw] = s + bias[2 * E + gw];
}

// ---------------------------------------------------------------------------
// out[o] = bout[o] + Wout[o] . vctx              (one wave per output)
__global__ void outK(const float* __restrict__ W, const float* __restrict__ bias,
                     const float* __restrict__ vctx, float* __restrict__ out) {
    int gw   = (blockIdx.x * blockDim.x + threadIdx.x) >> 5;
    int lane = threadIdx.x & 31;
    const float* wr = W + (size_t)gw * E;
    float s = 0.0f;
    for (int j = lane; j < E; j += 32) s += wr[j] * vctx[j];
    s = waveSum(s);
    if (lane == 0) out[gw] = s + bias[gw];
}

// ---------------------------------------------------------------------------
extern "C" void kernel_launch(void* const* d_in, const int* in_sizes, int n_in,
                              void* d_out, int out_size, void* d_ws, size_t ws_size,
                              hipStream_t stream) {
    const float* x    = (const float*)d_in[0];   // [L, E]
    const float* Win  = (const float*)d_in[1];   // [3E, E]
    const float* bin  = (const float*)d_in[2];   // [3E]
    const float* Wout = (const float*)d_in[3];   // [E, E]
    const float* bout = (const float*)d_in[4];   // [E]
    float* out = (float*)d_out;                  // [E]

    float* ws     = (float*)d_ws;
    float* q      = ws;                          // 1024
    float* consts = ws + 1024;                   // 16
    float* qkT    = ws + 2048;                   // 1024*16
    float* vctx   = ws + 18432;                  // 1024
    float* ctx    = ws + 19456;                  // 8*1024
    float* scores = ws + 27648;                  // 8*L
    float* attnT  = ws + 289792;                 // L*16   (end: 814080 floats)

    zeroK<<<(8 * E + 255) / 256, 256, 0, stream>>>(ctx, 8 * E);
    zeroK<<<(L * 16 + 255) / 256, 256, 0, stream>>>(attnT, L * 16);

    qprojK<<<128, 256, 0, stream>>>(x, Win, bin, q);
    qkK<<<4, 256, 0, stream>>>(Win, bin, q, qkT, consts);

    scoresK<<<L / 128, 256, 0, stream>>>(x, qkT, consts, scores);
    softmaxK<<<H, 256, 0, stream>>>(scores, attnT);
    ctxK<<<L / 128, 256, 0, stream>>>(x, attnT, ctx);

    vctxK<<<128, 256, 0, stream>>>(Win, bin, ctx, vctx);
    outK<<<128, 256, 0, stream>>>(Wout, bout, vctx, out);

    (void)in_sizes; (void)n_in; (void)out_size; (void)ws_size;
}